// B3SplineUWTApprox1_34797825032657
// MI455X (gfx1250) — compile-verified
//
#include <hip/hip_runtime.h>

typedef __attribute__((ext_vector_type(2))) float v2f;
typedef __attribute__((ext_vector_type(4))) float v4f;
typedef __attribute__((ext_vector_type(8))) float v8f;

#define ROWS    100
#define STRIDE  101     // odd -> conflict-free lane-strided LDS reads
#define HALO    14
#define TILE    64
#define REGION  92      // TILE + 2*HALO
#define NTHREADS 256
#define NWAVES   8

__device__ __forceinline__ int reflect_idx(int g, int n) {
    if (g < 0) g = -g;
    if (g >= n) g = 2 * n - 2 - g;
    return g;
}

// Async global->LDS copy of one dword per lane (ASYNCcnt-tracked).
__device__ __forceinline__ void async_copy_b32(const float* gsrc, float* ldst) {
#if __has_builtin(__builtin_amdgcn_global_load_async_to_lds_b32)
    __builtin_amdgcn_global_load_async_to_lds_b32(
        (__attribute__((address_space(1))) int*)(void*)gsrc,
        (__attribute__((address_space(3))) int*)(void*)ldst, 0, 0);
#else
    unsigned lds_off = (unsigned)(uintptr_t)ldst;
    asm volatile("global_load_async_to_lds_b32 %0, %1, off"
                 :: "v"(lds_off), "v"(gsrc) : "memory");
#endif
}

__device__ __forceinline__ void wait_asynccnt0() {
#if __has_builtin(__builtin_amdgcn_s_wait_asynccnt)
    __builtin_amdgcn_s_wait_asynccnt(0);
#else
    asm volatile("s_wait_asynccnt 0x0" ::: "memory");
#endif
}

__global__ __launch_bounds__(256) void uwt_fused_kernel(const float* __restrict__ x,
                                                        float* __restrict__ out) {
    __shared__ float lA[ROWS * STRIDE];   // current approximation y_j (region coords)
    __shared__ float lB[ROWS * STRIDE];   // horizontal-pass scratch

    const int tid  = threadIdx.x;
    const int lane = tid & 31;
    const int wave = tid >> 5;
    const int lm   = lane & 15;
    const int koff = (lane & 16) ? 2 : 0;   // K offset for upper half-wave (A/B frag layout)
    const int hi8  = (lane & 16) ? 8 : 0;   // M offset for upper half-wave (C/D layout)

    const int tileX = blockIdx.x;   // 16 tiles of 64 cols
    const int tileY = blockIdx.y;   // 16 tiles of 64 rows
    const int b     = blockIdx.z;   // 16 batches
    const int tr0 = tileY * TILE;
    const int tc0 = tileX * TILE;
    const int H = 1024, W = 1024;

    // Zero both LDS buffers: padding regions must be finite (0 * NaN = NaN in WMMA).
    for (int i = tid; i < ROWS * STRIDE; i += NTHREADS) { lA[i] = 0.f; lB[i] = 0.f; }
    __syncthreads();   // DScnt drained here -> zeros ordered before async LDS writes

    // Load 92x92 input region with reflect boundary handling via async global->LDS.
    const float* xb = x + (size_t)b * H * W;
    for (int i = tid; i < REGION * REGION; i += NTHREADS) {
        int rr = i / REGION, cc = i - rr * REGION;
        int gr = reflect_idx(tr0 - HALO + rr, H);
        int gc = reflect_idx(tc0 - HALO + cc, W);
        async_copy_b32(&xb[gr * W + gc], &lA[rr * STRIDE + cc]);
    }
    wait_asynccnt0();
    __syncthreads();

    const float W5[5] = {0.0625f, 0.25f, 0.375f, 0.25f, 0.0625f};
    const int dils[3]    = {1, 2, 4};
    const int log2d[3]   = {0, 1, 2};
    const int nchunks[3] = {5, 6, 8};      // K = 16 + 4*dil, in K4 chunks
    // pass1 spans the previous level's valid rows x next level's valid cols;
    // pass2 spans next level's valid rows x cols. (Overshoot tiles write only
    // into finite don't-care padding.)
    const int p1RowStart[3] = {0, 2, 6};
    const int p1RowCnt[3]   = {6, 6, 5};
    const int nxtStart[3]   = {2, 6, 14};
    const int nxtCnt[3]     = {6, 5, 4};

    float* outb = out + (size_t)b * 4 * H * W;

    for (int lev = 0; lev < 3; ++lev) {
        const int dil = dils[lev];
        const int nch = nchunks[lev];

        // Banded weight fragments. Per-lane values are identical whether used as
        // the B operand (pass1: W is KxN, value = w[i - n]) or the A operand
        // (pass2: W is MxK, value = w[i - m]) under the CDNA5 frag layouts.
        v2f wf[8];
        for (int c = 0; c < 8; ++c) {
            float w0 = 0.f, w1 = 0.f;
            if (c < nch) {
                int i0 = 4 * c + koff;
                int d0 = i0 - lm;
                if (d0 >= 0 && d0 <= 4 * dil && (d0 & (dil - 1)) == 0) w0 = W5[d0 >> log2d[lev]];
                int d1 = d0 + 1;
                if (d1 >= 0 && d1 <= 4 * dil && (d1 & (dil - 1)) == 0) w1 = W5[d1 >> log2d[lev]];
            }
            wf[c].x = w0; wf[c].y = w1;
        }

        // ---- pass 1 (horizontal): B[r][n] = sum_i A[r][n - 2*dil + i] * w(i-n)
        {
            const int nr = p1RowCnt[lev], ncl = nxtCnt[lev];
            const int rS = p1RowStart[lev], cS = nxtStart[lev];
            const int nt = nr * ncl;
            for (int t = wave; t < nt; t += NWAVES) {
                int ti = t / ncl, tj = t - ti * ncl;
                int r0 = rS + 16 * ti;
                int n0 = cS + 16 * tj;
                int cbase = n0 - 2 * dil;
                const float* arow = &lA[(r0 + lm) * STRIDE + cbase + koff];
                v8f acc = {};
                for (int c = 0; c < nch; ++c) {
                    v2f a;
                    a.x = arow[4 * c + 0];
                    a.y = arow[4 * c + 1];
                    acc = __builtin_amdgcn_wmma_f32_16x16x4_f32(
                        false, a, false, wf[c], (short)0, acc, false, false);
                }
                #pragma unroll
                for (int r = 0; r < 8; ++r)
                    lB[(r0 + r + hi8) * STRIDE + n0 + lm] = acc[r];
            }
        }
        __syncthreads();

        // ---- pass 2 (vertical): Anew[m][n] = sum_i w(i-m) * B[m - 2*dil + i][n]
        //      detail(lev) = Aold - Anew on the 64x64 center (streamed out NT)
        {
            const int ncl = nxtCnt[lev];
            const int S   = nxtStart[lev];
            const int nt  = ncl * ncl;
            for (int t = wave; t < nt; t += NWAVES) {
                int ti = t / ncl, tj = t - ti * ncl;
                int m0 = S + 16 * ti;
                int n0 = S + 16 * tj;
                int rbase = m0 - 2 * dil;
                const float* bcol = &lB[(rbase + koff) * STRIDE + n0 + lm];
                v8f acc = {};
                for (int c = 0; c < nch; ++c) {
                    v2f bb;
                    bb.x = bcol[(4 * c + 0) * STRIDE];
                    bb.y = bcol[(4 * c + 1) * STRIDE];
                    acc = __builtin_amdgcn_wmma_f32_16x16x4_f32(
                        false, wf[c], false, bb, (short)0, acc, false, false);
                }
                #pragma unroll
                for (int r = 0; r < 8; ++r) {
                    int m   = m0 + r + hi8;
                    int col = n0 + lm;
                    float oldv = lA[m * STRIDE + col];
                    float newv = acc[r];
                    lA[m * STRIDE + col] = newv;
                    if (m >= HALO && m < HALO + TILE && col >= HALO && col < HALO + TILE) {
                        int gr = tr0 + m - HALO;
                        int gc = tc0 + col - HALO;
                        __builtin_nontemporal_store(
                            oldv - newv, &outb[((size_t)lev * H + gr) * W + gc]);
                    }
                }
            }
        }
        __syncthreads();
    }

    // channel 3: final coarse approximation c_J (b128 NT stores, 16B-aligned)
    for (int i = tid; i < TILE * (TILE / 4); i += NTHREADS) {
        int r = i >> 4;            // row 0..63
        int c4 = (i & 15) * 4;     // col 0,4,...,60
        const float* src = &lA[(HALO + r) * STRIDE + HALO + c4];
        v4f v = {src[0], src[1], src[2], src[3]};
        v4f* dst = (v4f*)&outb[((size_t)3 * H + (tr0 + r)) * W + (tc0 + c4)];
        __builtin_nontemporal_store(v, dst);
    }
}

extern "C" void kernel_launch(void* const* d_in, const int* in_sizes, int n_in,
                              void* d_out, int out_size, void* d_ws, size_t ws_size,
                              hipStream_t stream) {
    const float* x = (const float*)d_in[0];   // (16, 1024, 1024) f32
    float* out = (float*)d_out;               // (16, 4, 1024, 1024) f32
    dim3 grid(16, 16, 16);                    // (tileX, tileY, batch)
    uwt_fused_kernel<<<grid, NTHREADS, 0, stream>>>(x, out);
}